// MSLoss_6992206758297
// MI455X (gfx1250) — compile-verified
//
#include <hip/hip_runtime.h>
#include <math.h>

// ---------------- geometry ----------------
#define PDIM  134                    // 128 + 2*3 pad
#define PD2   (PDIM*PDIM)            // 17956
#define NPAD  (PDIM*PDIM*PDIM)       // 2406104 per batch
#define N128  (128*128*128)          // 2097152 per batch
#define NBATCH 2
#define NBLK  1024                   // reduction blocks per batch

// ---------------- CDNA5 feature probes ----------------
#if defined(__has_builtin)
# if __has_builtin(__builtin_amdgcn_global_load_async_to_lds_b32)
#  define HAVE_ASYNC_LDS 1
# endif
# if __has_builtin(__builtin_amdgcn_s_wait_asynccnt)
#  define HAVE_WAIT_ASYNC 1
# endif
# if __has_builtin(__builtin_amdgcn_wmma_f32_16x16x4_f32)
#  define HAVE_WMMA_F32X4 1
# endif
#endif

typedef __attribute__((ext_vector_type(2))) float v2f;
typedef __attribute__((ext_vector_type(8))) float v8f;

#if defined(HAVE_ASYNC_LDS)
// Builtin params are AS1 (global) int* and AS3 (LDS) int* — reach the target
// address spaces via integer casts, pointee type int to match the prototype.
typedef __attribute__((address_space(1))) int* as1i_t;
typedef __attribute__((address_space(3))) int* as3i_t;
#endif

__device__ __forceinline__ void lds_copy_f32(const float* gsrc, float* ldst) {
#if defined(HAVE_ASYNC_LDS)
  __builtin_amdgcn_global_load_async_to_lds_b32(
      (as1i_t)(unsigned long long)gsrc,
      (as3i_t)(unsigned int)(unsigned long long)ldst, 0, 0);
#else
  *ldst = *gsrc;
#endif
}

__device__ __forceinline__ void wait_async_lds() {
#if defined(HAVE_ASYNC_LDS)
# if defined(HAVE_WAIT_ASYNC)
  __builtin_amdgcn_s_wait_asynccnt(0);
# else
  asm volatile("s_wait_asynccnt 0" ::: "memory");
# endif
#endif
}

// ---------------- kernel 1: sigmoid / threshold / pad ----------------
// X (padded) gets Vb with 1.0 border; VP gets sigmoid(Vlogit) (compact 128^3).
__global__ void prep_k(const float* __restrict__ vlogit,
                       float* __restrict__ vp, float* __restrict__ X) {
  long long i = (long long)blockIdx.x * blockDim.x + threadIdx.x;
  if (i >= (long long)NBATCH * NPAD) return;
  int b = (int)(i / NPAD);
  int r = (int)(i - (long long)b * NPAD);
  int z = r / PD2; int r2 = r - z * PD2; int y = r2 / PDIM; int x = r2 - y * PDIM;
  if (z < 3 || z >= PDIM - 3 || y < 3 || y >= PDIM - 3 || x < 3 || x >= PDIM - 3) {
    X[i] = 1.0f;
    return;
  }
  long long ii = ((long long)b * 128 + (z - 3)) * 128LL * 128LL
               + (long long)(y - 3) * 128 + (x - 3);
  float p = 1.0f / (1.0f + expf(-vlogit[ii]));
  vp[ii] = p;
  X[i] = (p > 0.7f) ? 1.0f : 0.0f;
}

// ---------------- kernel 2: erosion6 = 7-point min (OOB ignored) ----------------
__global__ void ero6_k(const float* __restrict__ in, float* __restrict__ out) {
  long long i = (long long)blockIdx.x * blockDim.x + threadIdx.x;
  if (i >= (long long)NBATCH * NPAD) return;
  int b = (int)(i / NPAD);
  int r = (int)(i - (long long)b * NPAD);
  int z = r / PD2; int r2 = r - z * PD2; int y = r2 / PDIM; int x = r2 - y * PDIM;
  const float* p = in + (long long)b * NPAD;
  float v = p[r];
  if (x > 0)        v = fminf(v, p[r - 1]);
  if (x < PDIM - 1) v = fminf(v, p[r + 1]);
  if (y > 0)        v = fminf(v, p[r - PDIM]);
  if (y < PDIM - 1) v = fminf(v, p[r + PDIM]);
  if (z > 0)        v = fminf(v, p[r - PD2]);
  if (z < PDIM - 1) v = fminf(v, p[r + PD2]);
  out[i] = v;
}

// ---------------- kernel 3: dilation26 (27-pt max, LDS tile) + fused epilogue --
// MODE 0: out = relu(e1 - max27(sten))                 (initial skeleton)
// MODE 1: out = e1 * max27(sten) + e2                  (skel reinforcement)
// MODE 2: d = relu(e1 - max27(sten)); s = e2; out = s + d - s*d   (combine)
// CLAMP:  out = min(out, 1) (where(out>1) semantics)
template <int MODE, int CLAMP>
__global__ void dil26_k(const float* __restrict__ sten,
                        const float* __restrict__ e1,
                        const float* __restrict__ e2,
                        float* __restrict__ out) {
  __shared__ float tile[1000];                 // 10x10x10 halo tile
  const int b  = blockIdx.z / 17;
  const int zt = blockIdx.z % 17;
  const int x0 = blockIdx.x * 8, y0 = blockIdx.y * 8, z0 = zt * 8;
  const float* sb = sten + (long long)b * NPAD;
  const int tid = (threadIdx.z * 8 + threadIdx.y) * 8 + threadIdx.x;

  for (int t = tid; t < 1000; t += 512) {
    int lz = t / 100; int rr = t - lz * 100; int ly = rr / 10; int lx = rr - ly * 10;
    int gz = z0 - 1 + lz, gy = y0 - 1 + ly, gx = x0 - 1 + lx;
    if (gz >= 0 && gz < PDIM && gy >= 0 && gy < PDIM && gx >= 0 && gx < PDIM) {
      lds_copy_f32(&sb[(long long)gz * PD2 + gy * PDIM + gx], &tile[t]);
    } else {
      tile[t] = -__builtin_inff();             // -inf: identity for max (SAME pad)
    }
  }
  wait_async_lds();
  __syncthreads();

  const int x = x0 + threadIdx.x, y = y0 + threadIdx.y, z = z0 + threadIdx.z;
  if (x >= PDIM || y >= PDIM || z >= PDIM) return;

  float m = -__builtin_inff();
#pragma unroll
  for (int dz = 0; dz < 3; ++dz)
#pragma unroll
    for (int dy = 0; dy < 3; ++dy)
#pragma unroll
      for (int dx = 0; dx < 3; ++dx)
        m = fmaxf(m, tile[((threadIdx.z + dz) * 10 + threadIdx.y + dy) * 10
                          + threadIdx.x + dx]);

  const long long o = (long long)b * NPAD + (long long)z * PD2 + y * PDIM + x;
  float r;
  if (MODE == 0) {
    r = fmaxf(e1[o] - m, 0.0f);
  } else if (MODE == 1) {
    r = e1[o] * m + e2[o];
  } else {
    float d = fmaxf(e1[o] - m, 0.0f);
    float s = e2[o];
    r = s + d - s * d;
  }
  if (CLAMP) r = (r > 1.0f) ? 1.0f : r;
  out[o] = r;
}

// ---------------- kernel 4: per-block partial sums (SoA output) ----------------
// Quantities: q0=sum(Vp*Vcl), q1=sum(Vcl), q2=sum(Vcl*S*Vp), q3=sum(S*Vp)
// red layout: red[q*(NBATCH*NBLK) + b*NBLK + blk] — contiguous plane per (q,b)
// so the WMMA finisher reads coalesced b32 per lane with zero lane shuffles.
__global__ void reduce1_k(const float* __restrict__ vp,
                          const float* __restrict__ vcl,
                          const float* __restrict__ S,
                          float* __restrict__ red) {
  const int b = blockIdx.y;
  const long long base = (long long)b * N128;
  float ax = 0.f, ay = 0.f, az = 0.f, aw = 0.f;
  for (int i = blockIdx.x * 256 + threadIdx.x; i < N128; i += NBLK * 256) {
    int x = i & 127, y = (i >> 7) & 127, z = (i >> 14) & 127;
    float p = vp[base + i];
    float c = vcl[base + i];
    float s = S[(long long)b * NPAD + (long long)(z + 3) * PD2
                + (y + 3) * PDIM + (x + 3)];
    float sp = s * p;
    ax += p * c; ay += c; az += c * sp; aw += sp;
  }
  __shared__ float4 sm[256];
  sm[threadIdx.x] = make_float4(ax, ay, az, aw);
  __syncthreads();
  for (int off = 128; off > 0; off >>= 1) {
    if ((int)threadIdx.x < off) {
      sm[threadIdx.x].x += sm[threadIdx.x + off].x;
      sm[threadIdx.x].y += sm[threadIdx.x + off].y;
      sm[threadIdx.x].z += sm[threadIdx.x + off].z;
      sm[threadIdx.x].w += sm[threadIdx.x + off].w;
    }
    __syncthreads();
  }
  if (threadIdx.x == 0) {
    const int slot = b * NBLK + blockIdx.x;
    red[0 * (NBATCH * NBLK) + slot] = sm[0].x;
    red[1 * (NBATCH * NBLK) + slot] = sm[0].y;
    red[2 * (NBATCH * NBLK) + slot] = sm[0].z;
    red[3 * (NBATCH * NBLK) + slot] = sm[0].w;
  }
}

// ---------------- kernel 5: one-wave WMMA finisher ----------------
// Sums 1024 f32 partials per (batch, quantity) with v_wmma_f32_16x16x4_f32
// against an all-ones B: D[m][n] = sum_k A[m][k] — one WMMA sums 64 f32 exactly.
__global__ void final_k(const float* __restrict__ red, float* __restrict__ out) {
  const int lane = threadIdx.x;               // 32 lanes, all active (EXEC all 1s)
  float tot[NBATCH][4];
#pragma unroll
  for (int b = 0; b < NBATCH; ++b) {
#pragma unroll
    for (int q = 0; q < 4; ++q) {
      const float* plane = red + (long long)q * (NBATCH * NBLK) + b * NBLK;
#if defined(HAVE_WMMA_F32X4)
      v8f acc = {0.f, 0.f, 0.f, 0.f, 0.f, 0.f, 0.f, 0.f};
      v2f ones; ones.x = 1.0f; ones.y = 1.0f;
      for (int i = 0; i < NBLK / 64; ++i) {
        v2f A;
        A.x = plane[i * 64 + lane];            // coalesced b32 per lane
        A.y = plane[i * 64 + 32 + lane];
        acc = __builtin_amdgcn_wmma_f32_16x16x4_f32(
            false, A, false, ones, (short)0, acc, false, false);
      }
      float s = acc[0] + acc[1] + acc[2] + acc[3] +
                acc[4] + acc[5] + acc[6] + acc[7];
      // column 0 of D: rows 0-7 live in lane 0, rows 8-15 in lane 16
      tot[b][q] = __shfl(s, 0, 32) + __shfl(s, 16, 32);
#else
      float s = 0.f;
      for (int e = lane; e < NBLK; e += 32) s += plane[e];
      for (int off = 16; off > 0; off >>= 1) s += __shfl_down(s, off, 32);
      tot[b][q] = __shfl(s, 0, 32);
#endif
    }
  }
  if (lane == 0) {
    const float eps = 1e-12f;
    float tp = 0.5f * ((tot[0][0] + eps) / (tot[0][1] + eps) +
                       (tot[1][0] + eps) / (tot[1][1] + eps));
    float ts = 0.5f * ((tot[0][2] + eps) / (tot[0][3] + eps) +
                       (tot[1][2] + eps) / (tot[1][3] + eps));
    out[0] = 1.0f - (2.0f * tp * ts + eps) / (tp + ts + eps);
  }
}

// ---------------- host orchestration ----------------
static void run_skeletonize(float*& Xb, float* SKb, float*& T1b, float*& T2b,
                            float* DDb, int iters, hipStream_t stream) {
  const long long totalP = (long long)NBATCH * NPAD;
  const int tpb = 256;
  const int nbP = (int)((totalP + tpb - 1) / tpb);
  dim3 g26(17, 17, 17 * NBATCH), b26(8, 8, 8);

  // skel = relu(X - dil26(ero6(X)))
  ero6_k<<<nbP, tpb, 0, stream>>>(Xb, T1b);
  dil26_k<0, 0><<<g26, b26, 0, stream>>>(T1b, Xb, Xb, SKb);

  for (int it = 0; it < iters; ++it) {
    const bool last = (it == iters - 1);
    ero6_k<<<nbP, tpb, 0, stream>>>(Xb, T2b);            // Xe
    ero6_k<<<nbP, tpb, 0, stream>>>(T2b, T1b);           // ero6(Xe)
    // DD = X * dil26(skel) + skel
    dil26_k<1, 0><<<g26, b26, 0, stream>>>(SKb, Xb, SKb, DDb);
    // skel = DD + delta - DD*delta,  delta = relu(Xe - dil26(ero6(Xe)))
    if (last) dil26_k<2, 1><<<g26, b26, 0, stream>>>(T1b, T2b, DDb, SKb);
    else      dil26_k<2, 0><<<g26, b26, 0, stream>>>(T1b, T2b, DDb, SKb);
    // X = Xe (pointer swap)
    float* t = Xb; Xb = T2b; T2b = t;
  }
}

extern "C" void kernel_launch(void* const* d_in, const int* in_sizes, int n_in,
                              void* d_out, int out_size, void* d_ws, size_t ws_size,
                              hipStream_t stream) {
  (void)in_sizes; (void)n_in; (void)out_size; (void)ws_size;
  const float* vlogit = (const float*)d_in[0];
  const float* vcl    = (const float*)d_in[1];
  // d_in[2] (Vedt) is unused by the reference loss.

  // workspace layout (floats)
  const long long PADN = (((long long)NBATCH * NPAD + 255) / 256) * 256;
  float* VP = (float*)d_ws;                 // 2 * 128^3
  float* B0 = VP + (long long)NBATCH * N128;
  float* X  = B0;
  float* SK = B0 + PADN;
  float* T1 = B0 + 2 * PADN;
  float* T2 = B0 + 3 * PADN;
  float* DD = B0 + 4 * PADN;
  float* RED = B0 + 5 * PADN;               // 4 * NBATCH * NBLK floats (SoA)

  const long long totalP = (long long)NBATCH * NPAD;
  const int tpb = 256;
  const int nbP = (int)((totalP + tpb - 1) / tpb);

  // 1) Vp = sigmoid(Vlogit); X = pad(Vb, 1.0)
  prep_k<<<nbP, tpb, 0, stream>>>(vlogit, VP, X);

  // 2) S = skeletonize(skeletonize(X, 5), 2)   (persistent skeletonization)
  run_skeletonize(X, SK, T1, T2, DD, 5, stream);   // result in SK
  float* X2 = SK;       // second pass input = first pass output
  float* SK2 = X;       // reuse X slot for second pass output (content dead)
  run_skeletonize(X2, SK2, T1, T2, DD, 2, stream); // result in SK2

  // 3) per-batch sums, then scalar loss
  reduce1_k<<<dim3(NBLK, NBATCH), 256, 0, stream>>>(VP, vcl, SK2, RED);
  final_k<<<1, 32, 0, stream>>>(RED, (float*)d_out);
}